// Loss_43198781063800
// MI455X (gfx1250) — compile-verified
//
#include <hip/hip_runtime.h>

typedef __attribute__((ext_vector_type(2))) float v2f;
typedef __attribute__((ext_vector_type(8))) float v8f;

#define WAVES_PER_BLOCK 8
#define ROWS_PER_WAVE   16

// For each of the 16 rows owned by a wave, find argmin_j ( ||ref_j||^2 - 2*q.r_j )
// over all M ref points, using V_WMMA_F32_16X16X4_F32 for the 16x16 dot tiles.
// Column-tile loop is unrolled x2 so two independent WMMA/epilogue chains
// overlap and hide the WMMA->VALU result latency.
__global__ __launch_bounds__(256) void nn_argmin_wmma(
    const float* __restrict__ query,  // [N,3]
    const float* __restrict__ ref,    // [M,3]
    int M,
    int* __restrict__ out_idx)        // [N]
{
    const int lane = threadIdx.x & 31;
    const int wave = threadIdx.x >> 5;
    const int half = lane >> 4;   // 0: K=0,1 lanes ; 1: K=2,3 lanes
    const int l16  = lane & 15;
    const int rowbase = (blockIdx.x * WAVES_PER_BLOCK + wave) * ROWS_PER_WAVE;

    // A tile: 16 query rows x K=4 (x,y,z,0). 2 VGPRs per lane, held all kernel.
    const int   qrow = rowbase + l16;
    const float qx = query[qrow * 3 + 0];
    const float qy = query[qrow * 3 + 1];
    const float qz = query[qrow * 3 + 2];
    v2f A;
    A.x = half ? qz : qx;
    A.y = half ? 0.0f : qy;

    float minv[8];
    int   mini[8];
#pragma unroll
    for (int r = 0; r < 8; ++r) { minv[r] = 3.4e38f; mini[r] = 0; }

    const int ntiles = M >> 4;          // assumes M % 32 == 0 (16384 here)
    for (int t = 0; t < ntiles; t += 2) {
        const int col0 = (t << 4) + l16;   // lane's column, tile t
        const int col1 = col0 + 16;        // lane's column, tile t+1

        const float r0x = ref[col0 * 3 + 0];
        const float r0y = ref[col0 * 3 + 1];
        const float r0z = ref[col0 * 3 + 2];
        const float r1x = ref[col1 * 3 + 0];
        const float r1y = ref[col1 * 3 + 1];
        const float r1z = ref[col1 * 3 + 2];

        // Branch-free streaming prefetch: both halves together cover the 32
        // columns consumed 8 iterations from now; clamp instead of branching.
        {
            int pc = col0 + 256 + (half << 4);
            pc = (pc < M) ? pc : col0;
            __builtin_prefetch(ref + pc * 3, 0, 3);
        }

        const float rr0 = r0x * r0x + r0y * r0y + r0z * r0z;
        const float rr1 = r1x * r1x + r1y * r1y + r1z * r1z;

        // B tiles: 4 x 16 (K x N), mirrored K-split: lane<16 -> (x,y), lane>=16 -> (z,0)
        v2f B0, B1;
        B0.x = half ? r0z : r0x;
        B0.y = half ? 0.0f : r0y;
        B1.x = half ? r1z : r1x;
        B1.y = half ? 0.0f : r1y;

        v8f acc0 = {};
        v8f acc1 = {};
        acc0 = __builtin_amdgcn_wmma_f32_16x16x4_f32(
            false, A, false, B0, (short)0, acc0, false, false);
        acc1 = __builtin_amdgcn_wmma_f32_16x16x4_f32(
            false, A, false, B1, (short)0, acc1, false, false);

        // acc[r] = dot(query[rowbase + r + half*8], ref[col]).
        // Tile t updated before tile t+1 so strict-< keeps first occurrence.
#pragma unroll
        for (int r = 0; r < 8; ++r) {
            const float c0 = __builtin_fmaf(-2.0f, acc0[r], rr0);
            if (c0 < minv[r]) { minv[r] = c0; mini[r] = col0; }
            const float c1 = __builtin_fmaf(-2.0f, acc1[r], rr1);
            if (c1 < minv[r]) { minv[r] = c1; mini[r] = col1; }
        }
    }

    // Reduce (min, idx) across each 16-lane half; first-occurrence tie-break.
#pragma unroll
    for (int r = 0; r < 8; ++r) {
        float v  = minv[r];
        int   ix = mini[r];
#pragma unroll
        for (int off = 8; off > 0; off >>= 1) {
            const float ov = __shfl_xor(v, off, 16);
            const int   oi = __shfl_xor(ix, off, 16);
            if (ov < v || (ov == v && oi < ix)) { v = ov; ix = oi; }
        }
        if (l16 == 0)
            out_idx[rowbase + half * 8 + r] = ix;
    }
}

// Exact squared-difference sums (matches reference: subtract actual points
// after the argmin). Deterministic per-block tree reduction -> partials.
__global__ __launch_bounds__(256) void exact_mse_partial(
    const float* __restrict__ a,    // [N,3]
    const float* __restrict__ b,    // [M,3]
    const int* __restrict__ idx,    // [N]
    int N,
    float* __restrict__ part)       // [gridDim.x]
{
    __shared__ float sdata[256];
    const int n = blockIdx.x * 256 + threadIdx.x;
    float s = 0.0f;
    if (n < N) {
        const int j = idx[n];
        const float dx = a[n * 3 + 0] - b[j * 3 + 0];
        const float dy = a[n * 3 + 1] - b[j * 3 + 1];
        const float dz = a[n * 3 + 2] - b[j * 3 + 2];
        s = dx * dx + dy * dy + dz * dz;
    }
    sdata[threadIdx.x] = s;
    __syncthreads();
    for (int st = 128; st > 0; st >>= 1) {
        if (threadIdx.x < st) sdata[threadIdx.x] += sdata[threadIdx.x + st];
        __syncthreads();
    }
    if (threadIdx.x == 0) part[blockIdx.x] = sdata[0];
}

__global__ __launch_bounds__(256) void finalize_loss(
    const float* __restrict__ partA, int nA,
    const float* __restrict__ partB, int nB,
    const float* __restrict__ weight,
    int N, int M,
    float* __restrict__ out)
{
    __shared__ float sA[256];
    __shared__ float sB[256];
    const int t = threadIdx.x;
    float a = 0.0f, b = 0.0f;
    for (int i = t; i < nA; i += 256) a += partA[i];
    for (int i = t; i < nB; i += 256) b += partB[i];
    sA[t] = a; sB[t] = b;
    __syncthreads();
    for (int st = 128; st > 0; st >>= 1) {
        if (t < st) { sA[t] += sA[t + st]; sB[t] += sB[t + st]; }
        __syncthreads();
    }
    if (t == 0) {
        const float w   = weight[0];
        const float B2A = sA[0] / (float)(N * 3);
        const float A2B = sB[0] / (float)(M * 3);
        out[0] = B2A * w + A2B * (1.0f - w);
    }
}

extern "C" void kernel_launch(void* const* d_in, const int* in_sizes, int n_in,
                              void* d_out, int out_size, void* d_ws, size_t ws_size,
                              hipStream_t stream)
{
    const float* pred = (const float*)d_in[0];  // [1,N,3]
    const float* gt   = (const float*)d_in[1];  // [1,M,3]
    const float* w    = (const float*)d_in[2];  // [1]

    const int N = in_sizes[0] / 3;
    const int M = in_sizes[1] / 3;

    char* ws = (char*)d_ws;
    int* idxA = (int*)ws;                              // N ints
    int* idxB = (int*)(ws + (size_t)N * sizeof(int));  // M ints
    const int nblkA = (N + 255) / 256;
    const int nblkB = (M + 255) / 256;
    float* partA = (float*)(ws + (size_t)(N + M) * sizeof(int));
    float* partB = partA + nblkA;

    // Direction 1: pred -> nearest gt
    nn_argmin_wmma<<<N / (ROWS_PER_WAVE * WAVES_PER_BLOCK), 256, 0, stream>>>(
        pred, gt, M, idxA);
    // Direction 2: gt -> nearest pred
    nn_argmin_wmma<<<M / (ROWS_PER_WAVE * WAVES_PER_BLOCK), 256, 0, stream>>>(
        gt, pred, N, idxB);

    exact_mse_partial<<<nblkA, 256, 0, stream>>>(pred, gt, idxA, N, partA);
    exact_mse_partial<<<nblkB, 256, 0, stream>>>(gt, pred, idxB, M, partB);

    finalize_loss<<<1, 256, 0, stream>>>(partA, nblkA, partB, nblkB, w, N, M,
                                         (float*)d_out);
}